// VoltageViolationLoss_75806172774903
// MI455X (gfx1250) — compile-verified
//
#include <hip/hip_runtime.h>

#define N_BUS     127
#define NROW      128
#define NCOL      16
#define NUM_CS    200
#define STATE_DIM 858
#define EV_START  258
#define ITERS     100
#define TOL       1e-6f

typedef float v2f __attribute__((ext_vector_type(2)));
typedef float v8f __attribute__((ext_vector_type(8)));

// ---------------------------------------------------------------------------
// Prologue: pack K (127x127 complex64, row-major) into a zero-padded,
// WMMA-A-fragment-friendly layout in d_ws:
//   Apack[(t*2 + half)*128 + row] = {Kr(row,k0), Kr(row,k0+1), Ki(row,k0), Ki(row,k0+1)}
// with k0 = 4t + 2*half.  Re-pair and Im-pair are register-adjacent so the
// b128 load destination sub-registers are directly usable as WMMA A operands.
// 8192 float4 = 128 KB.  OOB rows/cols -> 0.
// ---------------------------------------------------------------------------
__global__ __launch_bounds__(256)
void pack_k_kernel(const float2* __restrict__ K, float4* __restrict__ Apack) {
    int idx = blockIdx.x * 256 + threadIdx.x;      // 0..8191
    if (idx >= 32 * 2 * 128) return;
    int row  = idx & 127;
    int th   = idx >> 7;                           // 0..63
    int half = th & 1;
    int t    = th >> 1;
    int k0   = 4 * t + 2 * half;
    float4 v = make_float4(0.f, 0.f, 0.f, 0.f);
    if (row < N_BUS) {
        if (k0 < N_BUS)     { float2 c = K[row * N_BUS + k0];     v.x = c.x; v.z = c.y; }
        if (k0 + 1 < N_BUS) { float2 c = K[row * N_BUS + k0 + 1]; v.y = c.x; v.w = c.y; }
    }
    Apack[idx] = v;
}

// ---------------------------------------------------------------------------
// Main kernel: one block = 16 batch columns, 8 waves, wave w owns rows
// 16w..16w+15.  Complex fixed-point iteration v = K*conj(S/v) + Lm using
// V_WMMA_F32_16X16X4_F32; 4 independent accumulator chains per wave.
// ---------------------------------------------------------------------------
__global__ __launch_bounds__(256)
void pf_wmma_kernel(const float*  __restrict__ action,
                    const float*  __restrict__ state,
                    const float2* __restrict__ Lparam,
                    const float4* __restrict__ Apack,
                    float*        __restrict__ out) {
    __shared__ float SR [NROW * NCOL];   // Re(S)  [row][col]
    __shared__ float SI [NROW * NCOL];   // Im(S)
    __shared__ float LCR[NROW * NCOL];   // Re(Lc) [k][col]  (B-matrix staging)
    __shared__ float LCI[NROW * NCOL];   // Im(Lc)
    __shared__ float RED[8];
    __shared__ float OACC[NCOL];

    const int tid  = threadIdx.x;
    const int lane = tid & 31;
    const int w    = tid >> 5;           // wave id 0..7
    const int col  = lane & 15;          // N index (batch column within block)
    const int half = lane >> 4;          // lane half selects K-pair / row-half
    const int b0   = blockIdx.x * NCOL;

    // ---- front-end: EV power scatter + S assembly -------------------------
    for (int i = tid; i < NROW * NCOL; i += 256) SR[i] = 0.f;
    __syncthreads();

    if (tid < NCOL) {                    // one thread per batch column
        const float* st = state  + (size_t)(b0 + tid) * STATE_DIM;
        const float* ac = action + (size_t)(b0 + tid) * NUM_CS;
        for (int i = 0; i < NUM_CS; ++i) {
            float cap  = st[EV_START + 3 * i];
            int   bus  = (int)st[EV_START + 3 * i + 2];
            float conn = (cap > 0.f) ? 1.f : 0.f;
            float mch  = fminf(22.0f, conn * (70.0f - cap) * 4.0f);   // /0.25h
            float mdis = fmaxf(-22.0f, conn * (15.0f - cap) * 4.0f);
            float p    = 22.17f * ac[i];                              // MAX=|MIN|=22.17
            p = fmaxf(fminf(p, mch), mdis);
            SR[bus * NCOL + tid] += p;                                // serial per column
        }
    }
    __syncthreads();

    // finalize S = (p + iq)/1000 and init Lc = conj(S) (since v0 = 1)
    for (int i = tid; i < NROW * NCOL; i += 256) {
        int j = i >> 4, c = i & 15;
        float p = 0.f, q = 0.f;
        if (j < N_BUS) {
            const float* st = state + (size_t)(b0 + c) * STATE_DIM;
            p = (st[4 + j] + SR[i]) * 1e-3f;
            q =  st[4 + N_BUS + j]  * 1e-3f;
        }
        SR[i]  = p;  SI[i]  = q;
        LCR[i] = p;  LCI[i] = -q;
    }

    // Lm fragment (rows w*16 + half*8 + j), padded row -> 1+0i (harmless)
    v8f lmr, lmi, vmag;
    for (int j = 0; j < 8; ++j) {
        int row = w * 16 + half * 8 + j;
        float lr = 1.f, li = 0.f;
        if (row < N_BUS) { float2 L = Lparam[row]; lr = L.x; li = L.y; }
        lmr[j] = lr; lmi[j] = li;
        vmag[j] = 1.0f;                  // |v_init| = 1
    }
    __syncthreads();

    // ---- fixed-point iterations ------------------------------------------
    const int arow = w * 16 + col;       // this lane's A-matrix row
    bool done = false;
    for (int it = 0; it < ITERS; ++it) {
        if (done) break;

        // 4 independent accumulator chains: v_k = K@Lc + Lm
        v8f dr0 = lmr;                   // Kr*Lr
        v8f dr1 = {};                    // -Ki*Li
        v8f di0 = lmi;                   // Kr*Li
        v8f di1 = {};                    //  Ki*Lr
        #pragma unroll 4
        for (int t = 0; t < 32; ++t) {
            int    k0 = 4 * t + 2 * half;
            float4 af = Apack[(t * 2 + half) * 128 + arow];   // L2-hot b128
            v2f ar;  ar[0]  =  af.x; ar[1]  =  af.y;          // Kr pair (subregs)
            v2f ai;  ai[0]  =  af.z; ai[1]  =  af.w;          // Ki pair (subregs)
            v2f ain; ain[0] = -af.z; ain[1] = -af.w;          // -Ki (no NEG on f32 WMMA)
            v2f br;  br[0] = LCR[k0 * NCOL + col]; br[1] = LCR[(k0 + 1) * NCOL + col];
            v2f bi;  bi[0] = LCI[k0 * NCOL + col]; bi[1] = LCI[(k0 + 1) * NCOL + col];
            dr0 = __builtin_amdgcn_wmma_f32_16x16x4_f32(false, ar,  false, br, (short)0, dr0, false, false);
            dr1 = __builtin_amdgcn_wmma_f32_16x16x4_f32(false, ain, false, bi, (short)0, dr1, false, false);
            di0 = __builtin_amdgcn_wmma_f32_16x16x4_f32(false, ar,  false, bi, (short)0, di0, false, false);
            di1 = __builtin_amdgcn_wmma_f32_16x16x4_f32(false, ai,  false, br, (short)0, di1, false, false);
        }
        __syncthreads();                 // all waves done reading LCR/LCI

        // elementwise: tolerance, |v| update, new Lc = conj(S/v)
        float tolloc = 0.f;
        for (int j = 0; j < 8; ++j) {
            int   row = w * 16 + half * 8 + j;
            float vr = dr0[j] + dr1[j];
            float vi = di0[j] + di1[j];
            float d2 = vr * vr + vi * vi;
            float m  = sqrtf(d2);
            tolloc   = fmaxf(tolloc, fabsf(m - vmag[j]));
            vmag[j]  = m;
            int   idx = row * NCOL + col;
            float sr = SR[idx], si = SI[idx];
            float inv = 1.0f / d2;
            LCR[idx] = (sr * vr + si * vi) * inv;
            LCI[idx] = (sr * vi - si * vr) * inv;
        }

        // block-wide max reduction of tolerance (wave32 shuffle + LDS)
        for (int m = 16; m >= 1; m >>= 1)
            tolloc = fmaxf(tolloc, __shfl_xor(tolloc, m, 32));
        if (lane == 0) RED[w] = tolloc;
        __syncthreads();
        float bmax = RED[0];
        for (int r = 1; r < 8; ++r) bmax = fmaxf(bmax, RED[r]);
        done = (bmax < TOL);
    }

    // ---- loss: 1000 * sum_j min(0, 0.05 - |1 - |v||) per column ----------
    if (tid < NCOL) OACC[tid] = 0.f;
    __syncthreads();
    float lsum = 0.f;
    for (int j = 0; j < 8; ++j) {
        float dev = fabsf(1.0f - vmag[j]);      // padded row: |1-1| = 0
        lsum += fminf(0.0f, 0.05f - dev);
    }
    atomicAdd(&OACC[col], lsum);                // ds_add_f32
    __syncthreads();
    if (tid < NCOL) out[b0 + tid] = 1000.0f * OACC[tid];
}

// ---------------------------------------------------------------------------
extern "C" void kernel_launch(void* const* d_in, const int* in_sizes, int n_in,
                              void* d_out, int out_size, void* d_ws, size_t ws_size,
                              hipStream_t stream) {
    const float*  action = (const float*)d_in[0];   // (4096, 200) f32
    const float*  state  = (const float*)d_in[1];   // (4096, 858) f32
    const float2* K      = (const float2*)d_in[2];  // (127, 127) complex64
    const float2* L      = (const float2*)d_in[3];  // (127, 1)   complex64
    float4* Apack = (float4*)d_ws;                  // needs 128 KB scratch

    pack_k_kernel<<<32, 256, 0, stream>>>(K, Apack);
    pf_wmma_kernel<<<256, 256, 0, stream>>>(action, state, L, Apack, (float*)d_out);
}